// CrossModalAttention_43224550867416
// MI455X (gfx1250) — compile-verified
//
#include <hip/hip_runtime.h>

// CrossModalAttention — MI455X (gfx1250) implementation.
//
// The reference broadcasts `language` along T before the K/V projections, so
// K and V are T-replicated => softmax is exactly uniform (1/T, exact: T=2^10)
// => context = V row, independent of Q. Exact-equivalent computation:
//     f[b,:]   = (((language[b]@Wv+bv)@Wv2+bv2)@Wo+bo)@Wout+bout   (~19 MFLOP)
//     out[b,t] = state[b,t] + f[b]                                  (25 MB traffic)
// Stage 1: one workgroup (32 waves), fp32 WMMA 16x16x4, GEMM chain via LDS.
// Stage 2: bandwidth-bound float4 residual broadcast (~1.1 us at 23.3 TB/s).
//
// A-fragment trick: D rows 8..15 are discarded, and D row m depends only on
// A row m, so A rows 8..15 are filled with duplicates of rows 0..7
// (row = lane & 7). This makes every A load unconditional (no EXEC masking)
// and lets the K-pair load be a single 8-byte-aligned b64 load.

typedef __attribute__((ext_vector_type(2))) float v2f;
typedef __attribute__((ext_vector_type(8))) float v8f;

#define BATCH 8
#define TSEQ  1024
#define DDIM  384
#define DLANG 768
#define HDIM  512

// A-fragment (16x4 fp32, 2 VGPRs): lane m=L&15 holds row M=m; lanes 0-15 carry
// K=k0,k0+1 in v[0],v[1]; lanes 16-31 carry K=k0+2,k0+3. Rows 8..15 duplicate
// rows 0..7 (outputs for those rows are discarded).
__device__ __forceinline__ v2f load_a_frag(const float* __restrict__ Arow,
                                           int k0, int koff) {
  return *(const v2f*)(Arow + k0 + koff);   // 8B-aligned: k0%4==0, koff in {0,2}
}

// B-fragment (4x16 fp32, 2 VGPRs): lane n=L&15 holds column N=n0+n; lanes 0-15
// carry rows K=k0,k0+1 in v[0],v[1]; lanes 16-31 carry rows K=k0+2,k0+3.
__device__ __forceinline__ v2f load_b_frag(const float* __restrict__ Bcol,
                                           int ldb, int k0) {
  const float* p = Bcol + (size_t)k0 * ldb;
  v2f b;
  b.x = p[0];
  b.y = p[ldb];
  return b;
}

// One 16-wide N-tile per wave: out[0:8, n0:n0+16] = A[0:8,0:K] @ W[0:K,n0:n0+16] + bias.
// C/D layout (ISA 7.12.2): VGPR r, lanes 0-15 -> (M=r, N=n0+lane); lanes 16-31
// hold M=8..15 (duplicate rows, discarded).
__device__ __forceinline__ void wave_gemm(const float* __restrict__ A, int K,
                                          const float* __restrict__ W,
                                          const float* __restrict__ bias,
                                          float* __restrict__ out, int N,
                                          int wave, int lane, int nwaves) {
  const int koff        = (lane >> 4) << 1;        // 0 or 2: which K-pair this half-wave holds
  const float* Arow     = A + (lane & 7) * K;      // duplicated rows 8..15 -> rows 0..7
  for (int n0 = wave * 16; n0 < N; n0 += nwaves * 16) {
    const float* Bcol = W + (size_t)koff * N + n0 + (lane & 15);
    v8f acc = {0.f, 0.f, 0.f, 0.f, 0.f, 0.f, 0.f, 0.f};
#pragma unroll 8
    for (int k0 = 0; k0 < K; k0 += 4) {
      v2f a = load_a_frag(Arow, k0, koff);
      v2f b = load_b_frag(Bcol, N, k0);
      // D = A(16x4) x B(4x16) + C ; fp32 WMMA
      acc = __builtin_amdgcn_wmma_f32_16x16x4_f32(false, a, false, b,
                                                  (short)0, acc, false, false);
    }
    const float bb = bias[n0 + (lane & 15)];
    if (lane < 16) {   // only lanes 0-15 hold the valid rows M=0..7
#pragma unroll
      for (int r = 0; r < BATCH; ++r)
        out[r * N + n0 + lane] = acc[r] + bb;
    }
  }
}

__global__ void __launch_bounds__(1024)
lang_mlp_chain_kernel(const float* __restrict__ language,
                      const float* __restrict__ Wv,   const float* __restrict__ bv,
                      const float* __restrict__ Wv2,  const float* __restrict__ bv2,
                      const float* __restrict__ Wo,   const float* __restrict__ bo,
                      const float* __restrict__ Wout, const float* __restrict__ bout,
                      float* __restrict__ f_out /* [BATCH*DDIM] */) {
  __shared__ float v_in[BATCH * HDIM];   // language @ Wv + bv
  __shared__ float v2s [BATCH * HDIM];   // v_in @ Wv2 + bv2
  __shared__ float osh [BATCH * HDIM];   // v2 @ Wo + bo

  const int lane   = threadIdx.x & 31;
  const int wave   = threadIdx.x >> 5;
  const int nwaves = blockDim.x >> 5;    // 32 waves

  // GEMM1: [8,768] x [768,512]   (A straight from global, L2-resident)
  wave_gemm(language, DLANG, Wv, bv, v_in, HDIM, wave, lane, nwaves);
  __syncthreads();
  // GEMM2: [8,512] x [512,512]
  wave_gemm(v_in, HDIM, Wv2, bv2, v2s, HDIM, wave, lane, nwaves);
  __syncthreads();
  // GEMM3: [8,512] x [512,512]
  wave_gemm(v2s, HDIM, Wo, bo, osh, HDIM, wave, lane, nwaves);
  __syncthreads();
  // GEMM4: [8,512] x [512,384] -> global scratch
  wave_gemm(osh, HDIM, Wout, bout, f_out, DDIM, wave, lane, nwaves);
}

// out[b,t,:] = state[b,t,:] + f[b,:]  — pure HBM bandwidth, float4 path.
__global__ void __launch_bounds__(256)
residual_broadcast_kernel(const float4* __restrict__ state,
                          const float* __restrict__ f,
                          float4* __restrict__ out, int total4) {
  const int idx = blockIdx.x * blockDim.x + threadIdx.x;
  if (idx >= total4) return;
  const int D4 = DDIM / 4;               // 96 float4 per row
  const int d4 = idx % D4;
  const int b  = idx / (D4 * TSEQ);
  float4 s  = state[idx];
  float4 fv = reinterpret_cast<const float4*>(f)[b * D4 + d4];
  s.x += fv.x; s.y += fv.y; s.z += fv.z; s.w += fv.w;
  out[idx] = s;
}

extern "C" void kernel_launch(void* const* d_in, const int* in_sizes, int n_in,
                              void* d_out, int out_size, void* d_ws, size_t ws_size,
                              hipStream_t stream) {
  (void)in_sizes; (void)n_in; (void)out_size; (void)ws_size;
  const float* state    = (const float*)d_in[0];
  const float* language = (const float*)d_in[1];
  // d_in[2..5] = Wq,bq,Wk,bk ; d_in[8..11] = Wq2,bq2,Wk2,bk2 — provably unused
  const float* Wv   = (const float*)d_in[6];
  const float* bv   = (const float*)d_in[7];
  const float* Wv2  = (const float*)d_in[12];
  const float* bv2  = (const float*)d_in[13];
  const float* Wo   = (const float*)d_in[14];
  const float* bo   = (const float*)d_in[15];
  const float* Wout = (const float*)d_in[16];
  const float* bout = (const float*)d_in[17];

  float* f = (float*)d_ws;               // [8*384] fp32 scratch

  lang_mlp_chain_kernel<<<1, 1024, 0, stream>>>(language, Wv, bv, Wv2, bv2,
                                                Wo, bo, Wout, bout, f);

  const int total4 = (BATCH * TSEQ * DDIM) / 4;   // 786432
  residual_broadcast_kernel<<<(total4 + 255) / 256, 256, 0, stream>>>(
      (const float4*)state, f, (float4*)d_out, total4);
}